// FFFTrain_73169062855235
// MI455X (gfx1250) — compile-verified
//
#include <hip/hip_runtime.h>
#include <math.h>

// ---------------------------------------------------------------------------
// FFF (fast feedforward) forward, sparse path-walk formulation for MI455X.
//   x      : (16384, 1024) f32
//   w_in   : (4095, 1024)  f32   (row n = node n's input weights, contiguous)
//   w_out  : (1024, 4095)  f32   (transposed once into d_ws for coalescing)
//   out    : (16384, 1024) f32
// Per token only 12 tree nodes contribute. Logits computed with
// V_WMMA_F32_16X16X4_F32 on 16-token tiles (diagonal of X * gathered-W).
// ---------------------------------------------------------------------------

typedef float v2f __attribute__((ext_vector_type(2)));
typedef float v8f __attribute__((ext_vector_type(8)));

#define DEPTH     11
#define LEVELS    (DEPTH + 1)   // 12 path nodes per token (root + 11)
#define IN_W      1024
#define OUT_W     1024
#define N_NODES   4095
#define N_TOKENS  16384
#define TOK_TILE  16
#define XSTRIDE   1026          // padded LDS row stride (floats); even => b64-aligned

__device__ __forceinline__ v8f wmma4(v2f a, v2f b, v8f c) {
  // D(16x16,f32) = A(16x4,f32) * B(4x16,f32) + C
  return __builtin_amdgcn_wmma_f32_16x16x4_f32(
      /*neg_a=*/false, a, /*neg_b=*/false, b,
      /*c_mod=*/(short)0, c, /*reuse_a=*/false, /*reuse_b=*/false);
}

// ---------------------------------------------------------------------------
// Kernel 0: w_out (1024 x 4095) -> w_out_T (4095 x 1024), LDS-tiled transpose.
// ---------------------------------------------------------------------------
__global__ __launch_bounds__(256) void transpose_wout(const float* __restrict__ w_out,
                                                      float* __restrict__ w_out_T) {
  __shared__ float tile[32][33];
  const int n0 = blockIdx.x * 32;  // node dim (4095)
  const int o0 = blockIdx.y * 32;  // output dim (1024)
  const int tx = threadIdx.x;      // 0..31
  const int ty = threadIdx.y;      // 0..7
#pragma unroll
  for (int j = 0; j < 4; ++j) {
    const int n = n0 + tx;
    const int o = o0 + ty + j * 8;
    if (n < N_NODES) tile[ty + j * 8][tx] = w_out[(size_t)o * N_NODES + n];
  }
  __syncthreads();
#pragma unroll
  for (int j = 0; j < 4; ++j) {
    const int n = n0 + ty + j * 8;
    const int o = o0 + tx;
    if (n < N_NODES) w_out_T[(size_t)n * OUT_W + o] = tile[tx][ty + j * 8];
  }
}

// ---------------------------------------------------------------------------
// Kernel 1: per-tile tree walk + output accumulation.
// 128 threads (4 wave32) per block; block handles 16 tokens.
// Wave w owns K-slice [256w, 256w+256) of the dot products; partial diagonals
// are reduced across waves with ds_add_f32.
// ---------------------------------------------------------------------------
__global__ __launch_bounds__(128) void fff_fwd(const float* __restrict__ x,
                                               const float* __restrict__ w_in,
                                               const float* __restrict__ w_out_T,
                                               float* __restrict__ out) {
  __shared__ float lds_x[TOK_TILE * XSTRIDE];   // padded X tile
  __shared__ float s_logit[TOK_TILE];           // cross-wave logit reduction
  __shared__ int   s_node[TOK_TILE];            // current node per token
  __shared__ float s_act[LEVELS][TOK_TILE];     // gelu(logit) along the path
  __shared__ int   s_path[LEVELS][TOK_TILE];    // visited node ids

  const int tid  = threadIdx.x;
  const int lane = tid & 31;
  const int wave = tid >> 5;                    // 0..3
  const size_t xbase = (size_t)blockIdx.x * TOK_TILE * IN_W;

  // ---- stage 1: load X tile (coalesced b128) into padded LDS -------------
  const float4* __restrict__ xg = (const float4*)(x + xbase);
#pragma unroll 4
  for (int i = 0; i < 32; ++i) {
    const int idx = i * 128 + tid;              // 0..4095 float4s
    const int tok = idx >> 8;                   // 256 float4 per token row
    const int k4  = idx & 255;
    const float4 v = xg[idx];
    float* dst = &lds_x[tok * XSTRIDE + k4 * 4];
    dst[0] = v.x; dst[1] = v.y; dst[2] = v.z; dst[3] = v.w;
  }
  if (tid < TOK_TILE) { s_logit[tid] = 0.0f; s_node[tid] = 0; }
  __syncthreads();

  // WMMA f32 16x16x4 operand addressing: for both A(16x4) and B(4x16), lane l
  // serves token (l&15); upper half-lanes take K-offset +2. One b64 per lane.
  const int tok    = lane & 15;
  const int kShift = ((lane >> 4) & 1) * 2;
  const int kbase  = wave * 256;                // this wave's K slice
  const float* xrow = &lds_x[tok * XSTRIDE + kbase + kShift];

  // ---- stage 2: sequential tree walk, one WMMA-reduced level at a time ---
  for (int lvl = 0; lvl < LEVELS; ++lvl) {
    const int node = s_node[tok];
    const float* wrow = w_in + (size_t)node * IN_W + kbase + kShift;

    v8f c0 = {}, c1 = {}, c2 = {}, c3 = {};     // 4 accs break the C chain
#pragma unroll 4
    for (int kk = 0; kk < 256; kk += 16) {
      v2f a0 = *(const v2f*)(xrow + kk +  0);
      v2f b0 = *(const v2f*)(wrow + kk +  0);
      c0 = wmma4(a0, b0, c0);
      v2f a1 = *(const v2f*)(xrow + kk +  4);
      v2f b1 = *(const v2f*)(wrow + kk +  4);
      c1 = wmma4(a1, b1, c1);
      v2f a2 = *(const v2f*)(xrow + kk +  8);
      v2f b2 = *(const v2f*)(wrow + kk +  8);
      c2 = wmma4(a2, b2, c2);
      v2f a3 = *(const v2f*)(xrow + kk + 12);
      v2f b3 = *(const v2f*)(wrow + kk + 12);
      c3 = wmma4(a3, b3, c3);
    }
    const v8f c = (c0 + c1) + (c2 + c3);

    // Diagonal of C: lanes 0-7 hold C[t][t] (t=lane) at vgpr index = lane;
    // lanes 24-31 hold C[t][t] (t=lane-16) at vgpr index = lane-24.
    const int r = lane & 7;
    const float e0 = (r & 1) ? c[1] : c[0];
    const float e1 = (r & 1) ? c[3] : c[2];
    const float e2 = (r & 1) ? c[5] : c[4];
    const float e3 = (r & 1) ? c[7] : c[6];
    const float f0 = (r & 2) ? e1 : e0;
    const float f1 = (r & 2) ? e3 : e2;
    const float dg = (r & 4) ? f1 : f0;
    if (lane < 8)        atomicAdd(&s_logit[lane],      dg);  // ds_add_f32
    else if (lane >= 24) atomicAdd(&s_logit[lane - 16], dg);
    __syncthreads();

    if (tid < TOK_TILE) {                       // wave0 lanes 0-15: tok == tid
      const float logit = s_logit[tid];
      // exact GELU = 0.5*x*(1+erf(x/sqrt(2)))
      const float act = 0.5f * logit * (1.0f + erff(logit * 0.70710678118654752f));
      s_act[lvl][tid]  = act;
      s_path[lvl][tid] = node;
      s_node[tid]  = 2 * node + 1 + ((logit > 0.0f) ? 1 : 0);
      s_logit[tid] = 0.0f;
    }
    __syncthreads();
  }

  // ---- stage 3: out[t,:] = sum_k act_k * w_out_T[path_k, :] --------------
  // Wave w handles tokens 4w..4w+3; lane covers 4 contiguous floats x 8 chunks.
  float* orow_base = out + xbase;
  for (int t4 = 0; t4 < 4; ++t4) {
    const int t = wave * 4 + t4;
    float4 acc[8];
#pragma unroll
    for (int c8 = 0; c8 < 8; ++c8) acc[c8] = make_float4(0.f, 0.f, 0.f, 0.f);
    for (int k = 0; k < LEVELS; ++k) {
      const float a = s_act[k][t];
      const float4* row = (const float4*)(w_out_T + (size_t)s_path[k][t] * OUT_W);
#pragma unroll
      for (int c8 = 0; c8 < 8; ++c8) {
        const float4 w4 = row[c8 * 32 + lane];  // coalesced b128 from L2
        acc[c8].x += a * w4.x;
        acc[c8].y += a * w4.y;
        acc[c8].z += a * w4.z;
        acc[c8].w += a * w4.w;
      }
    }
    float4* orow = (float4*)(orow_base + (size_t)t * OUT_W);
#pragma unroll
    for (int c8 = 0; c8 < 8; ++c8) orow[c8 * 32 + lane] = acc[c8];
  }
}

// ---------------------------------------------------------------------------
extern "C" void kernel_launch(void* const* d_in, const int* in_sizes, int n_in,
                              void* d_out, int out_size, void* d_ws, size_t ws_size,
                              hipStream_t stream) {
  (void)in_sizes; (void)n_in; (void)out_size; (void)ws_size;
  const float* x     = (const float*)d_in[0];
  const float* w_in  = (const float*)d_in[1];
  const float* w_out = (const float*)d_in[2];
  // d_in[3] = force_depth (== DEPTH, compile-time constant here)
  float* w_out_T = (float*)d_ws;           // needs 4095*1024*4 = 16.8 MB scratch
  float* out     = (float*)d_out;

  transpose_wout<<<dim3((N_NODES + 31) / 32, OUT_W / 32), dim3(32, 8), 0, stream>>>(
      w_out, w_out_T);
  fff_fwd<<<N_TOKENS / TOK_TILE, 128, 0, stream>>>(x, w_in, w_out_T, out);
}